// GraphSAGE_29703993819225
// MI455X (gfx1250) — compile-verified
//
#include <hip/hip_runtime.h>
#include <hip/hip_bf16.h>
#include <math.h>

typedef __attribute__((ext_vector_type(2))) float v2f;
typedef __attribute__((ext_vector_type(8))) float v8f;

#define N_NODES 100000
#define N_EDGES 1600000
#define D_FEAT  128   // K dimension for every layer (D_IN == D_HID == 128)

// ---------------------------------------------------------------------------
// Degree: deg[dst] += 1
// ---------------------------------------------------------------------------
__global__ void degree_kernel(const long long* __restrict__ dst,
                              float* __restrict__ deg, int nEdges) {
    int e = blockIdx.x * blockDim.x + threadIdx.x;
    if (e < nEdges) {
        atomicAdd(&deg[(int)dst[e]], 1.0f);
    }
}

// deg -> 1/max(deg,1)   (in place)
__global__ void invdeg_kernel(float* __restrict__ deg, int n) {
    int i = blockIdx.x * blockDim.x + threadIdx.x;
    if (i < n) {
        deg[i] = 1.0f / fmaxf(deg[i], 1.0f);
    }
}

// ---------------------------------------------------------------------------
// Scatter-add: one wave per edge. Lane l gathers float4 l of the 128-float
// source row (one coalesced b128 load / lane = 512B per wave), then 4 f32
// atomic adds into the destination row.
// ---------------------------------------------------------------------------
__global__ void scatter_add_kernel(const long long* __restrict__ src,
                                   const long long* __restrict__ dst,
                                   const float* __restrict__ xin,
                                   float* __restrict__ agg, int nEdges) {
    int e = blockIdx.x * (blockDim.x >> 5) + (threadIdx.x >> 5);
    if (e >= nEdges) return;
    int lane = threadIdx.x & 31;
    int s = (int)src[e];
    int d = (int)dst[e];
    float4 v = ((const float4*)(xin + (size_t)s * D_FEAT))[lane];
    float* dp = agg + (size_t)d * D_FEAT + lane * 4;
    atomicAdd(dp + 0, v.x);
    atomicAdd(dp + 1, v.y);
    atomicAdd(dp + 2, v.z);
    atomicAdd(dp + 3, v.w);
}

// ---------------------------------------------------------------------------
// Pack W[128 x nOut] (row-major) into WMMA B-fragment order:
//   P[((c*32 + t)*32 + lane)] = { W[4t+2h][16c+lr], W[4t+2h+1][16c+lr] }
// so the GEMM fetches each B fragment with a single coalesced b64 load.
// ---------------------------------------------------------------------------
__global__ void pack_weights_kernel(const float* __restrict__ W,
                                    float* __restrict__ P, int nOut) {
    int t = blockIdx.x * blockDim.x + threadIdx.x;
    int total = (nOut >> 4) * 32 * 32;
    if (t >= total) return;
    int lane  = t & 31;
    int kstep = (t >> 5) & 31;
    int c     = t >> 10;
    int half = lane >> 4, lr = lane & 15;
    int ka  = kstep * 4 + 2 * half;
    int col = c * 16 + lr;
    P[(size_t)t * 2 + 0] = W[(size_t)ka * nOut + col];
    P[(size_t)t * 2 + 1] = W[(size_t)(ka + 1) * nOut + col];
}

// ---------------------------------------------------------------------------
// SAGE linear:  out = (agg * invdeg) @ Wl + bias + xin @ Wr   [+ ReLU]
//
// Block = ROWTILES row-tiles x COLT column-tiles (one wave per 16x16 tile).
// A tiles (mean-scaled agg, and x) are staged in LDS once per block; B comes
// from the packed weight buffers as coalesced b64 loads. Full-fp32 WMMA
// (V_WMMA_F32_16X16X4_F32), 2 per K-step, K-loop of 32 steps.
//
// Fragment layouts (CDNA5 ISA 7.12.2, wave32):
//   A 16x4:  lane l (lr=l&15, h=l>>4): {A[lr][k0+2h], A[lr][k0+2h+1]}
//   B 4x16:  lane l:                   {B[k0+2h][lr], B[k0+2h+1][lr]}
//   C/D 16x16: acc[m] = D[m + 8*h][lr]
// ---------------------------------------------------------------------------
template <int NOUT, int ROWTILES>
__global__ void sage_gemm_kernel(const float* __restrict__ agg,
                                 const float* __restrict__ invdeg,
                                 const float* __restrict__ xin,
                                 const float* __restrict__ pWl,
                                 const float* __restrict__ bias,
                                 const float* __restrict__ pWr,
                                 float* __restrict__ out,
                                 int doRelu) {
    constexpr int COLT = NOUT / 16;
    constexpr int ROWS = 16 * ROWTILES;
    constexpr int STR  = 130;              // padded row stride (bank spread)
    __shared__ float sAgg[ROWS * STR];
    __shared__ float sX[ROWS * STR];

    const int nodeBase = blockIdx.x * ROWS;

    // Cooperative stage: ROWS x 128 floats per matrix, float4 granularity.
    for (int i = threadIdx.x; i < ROWS * 32; i += blockDim.x) {
        int r  = i >> 5;
        int c4 = i & 31;
        int node = nodeBase + r;
        if (node < N_NODES) {
            float sc = invdeg[node];
            float4 va = ((const float4*)(agg + (size_t)node * D_FEAT))[c4];
            float4 vx = ((const float4*)(xin + (size_t)node * D_FEAT))[c4];
            float* pa = &sAgg[r * STR + c4 * 4];
            float* px = &sX[r * STR + c4 * 4];
            pa[0] = va.x * sc; pa[1] = va.y * sc; pa[2] = va.z * sc; pa[3] = va.w * sc;
            px[0] = vx.x;      px[1] = vx.y;      px[2] = vx.z;      px[3] = vx.w;
        }
    }
    __syncthreads();

    const int w       = threadIdx.x >> 5;
    const int rowSub  = w / COLT;
    const int colTile = w % COLT;
    const int row0    = nodeBase + rowSub * 16;
    if (row0 >= N_NODES) return;           // wave-uniform exit (after barrier)

    const int lane = threadIdx.x & 31;
    const int half = lane >> 4;
    const int lr   = lane & 15;

    const v2f* pL = (const v2f*)pWl + (size_t)colTile * 32 * 32 + lane;
    const v2f* pR = (const v2f*)pWr + (size_t)colTile * 32 * 32 + lane;
    const float* aRow = &sAgg[(rowSub * 16 + lr) * STR];
    const float* xRow = &sX[(rowSub * 16 + lr) * STR];

    v8f acc = {};
    #pragma unroll 8
    for (int t = 0; t < 32; ++t) {
        const int ka = 4 * t + 2 * half;
        v2f aA = *(const v2f*)&aRow[ka];   // ds_load_b64
        v2f bL = pL[t * 32];               // coalesced global_load_b64
        acc = __builtin_amdgcn_wmma_f32_16x16x4_f32(
                  false, aA, false, bL, (short)0, acc, false, false);
        v2f aX = *(const v2f*)&xRow[ka];
        v2f bR = pR[t * 32];
        acc = __builtin_amdgcn_wmma_f32_16x16x4_f32(
                  false, aX, false, bR, (short)0, acc, false, false);
    }

    const float bv = bias[colTile * 16 + lr];
    #pragma unroll
    for (int m = 0; m < 8; ++m) {
        float v = acc[m] + bv;
        if (doRelu) v = fmaxf(v, 0.0f);
        out[(size_t)(row0 + m + 8 * half) * NOUT + colTile * 16 + lr] = v;
    }
}

// ---------------------------------------------------------------------------
// log_softmax over 32 classes: one wave per node, one lane per class.
// ---------------------------------------------------------------------------
__global__ void log_softmax32_kernel(float* __restrict__ out, int n) {
    int wave = (blockIdx.x * blockDim.x + threadIdx.x) >> 5;
    int lane = threadIdx.x & 31;
    if (wave >= n) return;
    size_t idx = (size_t)wave * 32 + lane;
    float v = out[idx];
    float m = v;
    #pragma unroll
    for (int off = 16; off >= 1; off >>= 1)
        m = fmaxf(m, __shfl_xor(m, off, 32));
    float e = expf(v - m);
    float s = e;
    #pragma unroll
    for (int off = 16; off >= 1; off >>= 1)
        s += __shfl_xor(s, off, 32);
    out[idx] = (v - m) - logf(s);
}

// ---------------------------------------------------------------------------
// Host orchestration
// ---------------------------------------------------------------------------
extern "C" void kernel_launch(void* const* d_in, const int* in_sizes, int n_in,
                              void* d_out, int out_size, void* d_ws, size_t ws_size,
                              hipStream_t stream) {
    (void)in_sizes; (void)n_in; (void)out_size; (void)ws_size;

    const float*     x   = (const float*)d_in[0];
    const long long* ei  = (const long long*)d_in[1];   // (2, E) int64
    const float*     Wl0 = (const float*)d_in[2];
    const float*     bl0 = (const float*)d_in[3];
    const float*     Wr0 = (const float*)d_in[4];
    const float*     Wl1 = (const float*)d_in[5];
    const float*     bl1 = (const float*)d_in[6];
    const float*     Wr1 = (const float*)d_in[7];
    const float*     Wl2 = (const float*)d_in[8];
    const float*     bl2 = (const float*)d_in[9];
    const float*     Wr2 = (const float*)d_in[10];
    float*           out = (float*)d_out;

    const long long* src = ei;             // edge_index[0]
    const long long* dst = ei + N_EDGES;   // edge_index[1]

    // Workspace layout (floats): deg | agg | h1 | h2 | 6 packed weight bufs
    const size_t degPad = 102400;
    const size_t featSz = (size_t)N_NODES * D_FEAT;
    float* deg  = (float*)d_ws;
    float* agg  = deg + degPad;
    float* h1   = agg + featSz;
    float* h2   = h1 + featSz;
    float* pWl0 = h2   + featSz;           // 128*128 floats each
    float* pWr0 = pWl0 + 128 * 128;
    float* pWl1 = pWr0 + 128 * 128;
    float* pWr1 = pWl1 + 128 * 128;
    float* pWl2 = pWr1 + 128 * 128;        // 128*32 floats each
    float* pWr2 = pWl2 + 128 * 32;

    const int B = 256;

    // One-time (per launch) weight packing into WMMA fragment order.
    pack_weights_kernel<<<(8 * 1024 + B - 1) / B, B, 0, stream>>>(Wl0, pWl0, 128);
    pack_weights_kernel<<<(8 * 1024 + B - 1) / B, B, 0, stream>>>(Wr0, pWr0, 128);
    pack_weights_kernel<<<(8 * 1024 + B - 1) / B, B, 0, stream>>>(Wl1, pWl1, 128);
    pack_weights_kernel<<<(8 * 1024 + B - 1) / B, B, 0, stream>>>(Wr1, pWr1, 128);
    pack_weights_kernel<<<(2 * 1024 + B - 1) / B, B, 0, stream>>>(Wl2, pWl2, 32);
    pack_weights_kernel<<<(2 * 1024 + B - 1) / B, B, 0, stream>>>(Wr2, pWr2, 32);

    // Degree + inverse degree (reused by all 3 layers)
    hipMemsetAsync(deg, 0, (size_t)N_NODES * sizeof(float), stream);
    degree_kernel<<<(N_EDGES + B - 1) / B, B, 0, stream>>>(dst, deg, N_EDGES);
    invdeg_kernel<<<(N_NODES + B - 1) / B, B, 0, stream>>>(deg, N_NODES);

    const int scatterBlocks = (N_EDGES + (B / 32) - 1) / (B / 32);  // 1 wave/edge
    const int gemmBlocks128 = N_NODES / 16;    // 6250, blockDim 256 (8 waves)
    const int gemmBlocks32  = N_NODES / 32;    // 3125, blockDim 128 (4 waves)

    // Layer 0
    hipMemsetAsync(agg, 0, featSz * sizeof(float), stream);
    scatter_add_kernel<<<scatterBlocks, B, 0, stream>>>(src, dst, x, agg, N_EDGES);
    sage_gemm_kernel<128, 1><<<gemmBlocks128, 256, 0, stream>>>(
        agg, deg, x, pWl0, bl0, pWr0, h1, 1);
    // Layer 1
    hipMemsetAsync(agg, 0, featSz * sizeof(float), stream);
    scatter_add_kernel<<<scatterBlocks, B, 0, stream>>>(src, dst, h1, agg, N_EDGES);
    sage_gemm_kernel<128, 1><<<gemmBlocks128, 256, 0, stream>>>(
        agg, deg, h1, pWl1, bl1, pWr1, h2, 1);
    // Layer 2 (straight into d_out, no ReLU)
    hipMemsetAsync(agg, 0, featSz * sizeof(float), stream);
    scatter_add_kernel<<<scatterBlocks, B, 0, stream>>>(src, dst, h2, agg, N_EDGES);
    sage_gemm_kernel<32, 2><<<gemmBlocks32, 128, 0, stream>>>(
        agg, deg, h2, pWl2, bl2, pWr2, out, 0);

    // log_softmax in place on d_out
    log_softmax32_kernel<<<(N_NODES * 32 + B - 1) / B, B, 0, stream>>>(out, N_NODES);
}